// Gather2D_7687991459873
// MI455X (gfx1250) — compile-verified
//
#include <hip/hip_runtime.h>

// Gather2D: out[i][c] = image[c*H*W + clamp(x_i)*W + clamp(y_i)]
// image: (1, 256, 512, 512) f32; coords: (N, 2) int32; out: (N, 256) f32.
//
// Memory-bound gather. Strategy:
//  - gridDim.y = 8 channel chunks of 32 channels -> live image working set
//    ~32 MB, resident in the 192 MB L2, so each 1 MB plane streams from HBM
//    exactly once while servicing ~100k random lookups from L2.
//  - 256-thread block = 32 coords x 32 channels; each thread does 4 scalar
//    plane loads (1 MB apart, folded into global_load_b32 immediate offsets)
//    and one 128-bit nontemporal store; 8 adjacent lanes produce one full
//    128 B output cacheline -> perfect write combining, no L2 pollution.

typedef float __attribute__((ext_vector_type(4))) floatx4;

__global__ __launch_bounds__(256) void Gather2D_kernel(
    const float* __restrict__ image,
    const int*   __restrict__ coords,   // (N,2) int32
    float*       __restrict__ out,      // (N,256) f32
    int n)
{
    constexpr int CH    = 256;
    constexpr int Hdim  = 512;
    constexpr int Wdim  = 512;
    constexpr int PLANE = Hdim * Wdim;  // 262144 elements = 1 MB

    const int t        = threadIdx.x;
    const int cSlot    = t & 7;         // 0..7  -> 4 channels each
    const int coordSub = t >> 3;        // 0..31 -> coord within block
    const int i        = blockIdx.x * 32 + coordSub;
    if (i >= n) return;

    const int cBase = blockIdx.y * 32 + cSlot * 4;   // 0..252, step 4

    // One coalesced 8-byte load per coord (8 lanes share the same line).
    const int2 xy = ((const int2*)coords)[i];
    const int x = min(max(xy.x, 0), Hdim - 1);
    const int y = min(max(xy.y, 0), Wdim - 1);
    const size_t pos = (size_t)(x * Wdim + y);

    // Base address for this thread's 4 planes; the 3 extra planes are
    // reached via 1 MB / 2 MB / 3 MB immediate offsets (fits IOFFSET i24).
    const float* p = image + (size_t)cBase * PLANE + pos;
    floatx4 v;
    v.x = p[0 * PLANE];
    v.y = p[1 * PLANE];
    v.z = p[2 * PLANE];
    v.w = p[3 * PLANE];

    // 16 B aligned (cBase % 4 == 0, row length 256 floats).
    floatx4* dst = (floatx4*)(out + (size_t)i * CH + cBase);
    __builtin_nontemporal_store(v, dst);
}

extern "C" void kernel_launch(void* const* d_in, const int* in_sizes, int n_in,
                              void* d_out, int out_size, void* d_ws, size_t ws_size,
                              hipStream_t stream) {
    const float* image  = (const float*)d_in[0];
    const int*   coords = (const int*)d_in[1];
    float*       out    = (float*)d_out;

    const int n = in_sizes[1] / 2;   // (N,2) -> N coords

    dim3 block(256, 1, 1);
    dim3 grid((n + 31) / 32, 8, 1);  // y: 8 chunks of 32 channels
    Gather2D_kernel<<<grid, block, 0, stream>>>(image, coords, out, n);
}